// ImageFeatureModule_72155450573442
// MI455X (gfx1250) — compile-verified
//
#include <hip/hip_runtime.h>

typedef __attribute__((ext_vector_type(2))) float v2f;
typedef __attribute__((ext_vector_type(8))) float v8f;

#define MAX_IMVOTE 3
#define VD 13

// One wave32 handles 32 points via two V_WMMA_F32_16X16X4_F32 tiles.
// A (16x4 f32, 2 VGPRs): lanes 0-15 hold K=0 (v0) / K=1 (v1) for M=0..15,
//                        lanes 16-31 hold K=2 (v0) / K=3 (v1) for M=0..15.
// B (4x16 f32, 2 VGPRs): v0 lanes 0-15 = row K=0, lanes 16-31 = row K=2;
//                        v1 lanes 0-15 = row K=1, lanes 16-31 = row K=3 (zero).
// D (16x16 f32, 8 VGPRs): element (M = r + 8*(lane>=16), N = lane%16).
__global__ __launch_bounds__(256)
void imvote_feat_kernel(const float* __restrict__ Rtilt,
                        const float* __restrict__ scale,
                        const float* __restrict__ xyz,
                        const float* __restrict__ Kc,
                        const float* __restrict__ votes,
                        const int*   __restrict__ width,
                        float* __restrict__ out,
                        int B, int N, long votesPerBatch)
{
    // 8 waves/block, per-wave 2 tiles of 16 rows x 4 useful cols
    __shared__ __align__(16) float lds[8][2][16][4];

    const int  tid  = threadIdx.x;
    const int  lane = tid & 31;
    const int  wv   = tid >> 5;
    const long wavePt0 = (long)blockIdx.x * 256 + (long)wv * 32;
    const int  b = (int)(wavePt0 / N);   // N multiple of 256 -> batch uniform per block

    // ---------- per-batch fused 4x3 matrix ----------
    const float* R = Rtilt + b * 9;
    const float* K = Kc    + b * 9;
    const float  s   = scale[b];
    const float  is2 = 1.0f / (s * s);

    // xyz_cam = mcam @ xyz  (Rtilt^T, axis swap (x,z,y), negate new y, 1/s^2)
    float mcam[3][3];
    #pragma unroll
    for (int k = 0; k < 3; ++k) {
        mcam[0][k] =  is2 * R[k * 3 + 0];
        mcam[1][k] = -is2 * R[k * 3 + 2];
        mcam[2][k] =  is2 * R[k * 3 + 1];
    }
    // rows 0..2: K @ mcam (-> uv homogeneous), row 3: mcam row 2 (-> cam z)
    float Mc[4][3];
    #pragma unroll
    for (int r = 0; r < 3; ++r)
        #pragma unroll
        for (int k = 0; k < 3; ++k)
            Mc[r][k] = K[r*3+0]*mcam[0][k] + K[r*3+1]*mcam[1][k] + K[r*3+2]*mcam[2][k];
    #pragma unroll
    for (int k = 0; k < 3; ++k) Mc[3][k] = mcam[2][k];

    // ---------- B operand: B[k][n] = Mc[n][k] ----------
    const int  nn = lane & 15;
    const bool hi = lane >= 16;
    v2f bm; bm.x = 0.0f; bm.y = 0.0f;
    if (nn < 4) {
        bm.x = hi ? Mc[nn][2] : Mc[nn][0];
        bm.y = hi ? 0.0f      : Mc[nn][1];   // K=3 row is zero
    }

    // ---------- A operands (two 16-point tiles) ----------
    const float* x0 = xyz + (wavePt0 + nn) * 3;
    const float* x1 = xyz + (wavePt0 + 16 + nn) * 3;
    v2f a0, a1;
    a0.x = hi ? x0[2] : x0[0];
    a0.y = hi ? 0.0f  : x0[1];
    a1.x = hi ? x1[2] : x1[0];
    a1.y = hi ? 0.0f  : x1[1];

    v8f cz = {0.f, 0.f, 0.f, 0.f, 0.f, 0.f, 0.f, 0.f};
    v8f d0 = __builtin_amdgcn_wmma_f32_16x16x4_f32(false, a0, false, bm, (short)0, cz, false, false);
    v8f d1 = __builtin_amdgcn_wmma_f32_16x16x4_f32(false, a1, false, bm, (short)0, cz, false, false);

    // ---------- redistribute D columns 0..3 to one-lane-per-point ----------
    const int rowBase = hi ? 8 : 0;
    if (nn < 4) {
        #pragma unroll
        for (int r = 0; r < 8; ++r) {
            lds[wv][0][rowBase + r][nn] = d0[r];
            lds[wv][1][rowBase + r][nn] = d1[r];
        }
    }
    // same-wave DS ops are pipeline-ordered; compiler inserts the dscnt wait
    const int    tile = lane >> 4;
    const float4 rr   = *(const float4*)&lds[wv][tile][nn][0];

    // ---------- per-point epilogue: project, gather, emit ----------
    const float uvw  = rr.z;                       // uv homogeneous w (depth)
    const float camz = rr.w;                       // xyz_cam z
    const float u    = rintf(rr.x / uvw - 1.0f);   // jnp.round = RNE
    const float v    = rintf(rr.y / uvw - 1.0f);
    const float wpx  = (float)width[b];
    const int   idx_beg = (int)((u + v * wpx) * (float)VD); // astype(int32): trunc

    const float* vb  = votes + (long)b * votesPerBatch;
    const float  cnt = vb[idx_beg];
    const float  sc  = camz / K[0];                // xyz_cam_z / focal

    const long p = wavePt0 + lane;
    const long tensorStride = (long)B * N * 6;
    #pragma unroll
    for (int i = 0; i < MAX_IMVOTE; ++i) {
        const float g0 = vb[idx_beg + 1 + 4 * i];
        const float g1 = vb[idx_beg + 2 + 4 * i];
        const float g2 = vb[idx_beg + 4 + 4 * i];
        const float m  = (cnt > (float)i) ? 1.0f : 0.0f;
        float* o = out + (long)i * tensorStride + p * 6;
        o[0] = g0 * sc;
        o[1] = g1 * sc;
        o[2] = u;
        o[3] = v;
        o[4] = g2;
        o[5] = m;
    }
}

extern "C" void kernel_launch(void* const* d_in, const int* in_sizes, int n_in,
                              void* d_out, int out_size, void* d_ws, size_t ws_size,
                              hipStream_t stream) {
    const float* Rtilt = (const float*)d_in[0];   // [B,3,3]
    const float* scale = (const float*)d_in[1];   // [B]
    const float* xyz   = (const float*)d_in[2];   // [B,N,3]
    const float* K     = (const float*)d_in[3];   // [B,3,3]
    const float* votes = (const float*)d_in[4];   // [B, W*H*13]
    const int*   width = (const int*)d_in[5];     // [B]
    float* out = (float*)d_out;

    const int  B = in_sizes[1];
    const int  N = in_sizes[2] / (3 * B);
    const long votesPerBatch = (long)in_sizes[4] / B;

    const int totalPts = B * N;        // 16384; multiple of 256 by construction
    const int blocks   = totalPts / 256;
    imvote_feat_kernel<<<blocks, 256, 0, stream>>>(
        Rtilt, scale, xyz, K, votes, width, out, B, N, votesPerBatch);
}